// GAT_66709432041923
// MI455X (gfx1250) — compile-verified
//
#include <hip/hip_runtime.h>
#include <math.h>

#define N_NODES   50000
#define N_EDGES   800000
#define F_IN      128
#define HID_D     64
#define H1_HEADS  8
#define OUT_D     64
#define NEG_SLOPE 0.2f

typedef __attribute__((ext_vector_type(2))) float v2f;
typedef __attribute__((ext_vector_type(8))) float v8f;

// ---------- monotonic float<->uint encoding for atomicMax on floats ----------
__device__ __forceinline__ unsigned fenc(float f) {
    unsigned u = __float_as_uint(f);
    return (u & 0x80000000u) ? ~u : (u | 0x80000000u);
}
__device__ __forceinline__ float fdec(unsigned u) {
    unsigned v = (u & 0x80000000u) ? (u ^ 0x80000000u) : ~u;
    return __uint_as_float(v);
}
#define ENC_NEG_INF 0x007FFFFFu   // fenc(-inf)

// ---------- fp32 WMMA GEMM: C[M,N] = A[M,K] * B[K,N] ----------
// One 16x16 tile per wave; 4 waves per block cover 4 consecutive N-tiles.
// grid = (M/16, N/64), block = 128.  M,K,N multiples of 16/4 (guaranteed here).
__global__ void gemm_wmma_f32(const float* __restrict__ A,
                              const float* __restrict__ B,
                              float* __restrict__ C, int K, int N) {
    const int lane  = threadIdx.x & 31;
    const int wave  = threadIdx.x >> 5;
    const int mtile = blockIdx.x;
    const int ntile = blockIdx.y * 4 + wave;
    const int ml    = lane & 15;     // matrix row (A) / col (B,C) within tile
    const int hi    = lane >> 4;     // half-wave: selects K pair / row half

    // A 16x4 layout: lanes 0-15 -> K=0,1 ; lanes 16-31 -> K=2,3
    const float* arow = A + (size_t)(mtile * 16 + ml) * K + hi * 2;
    // B 4x16 layout: vgpr0 = K {0 | 2}, vgpr1 = K {1 | 3}, col = lane&15
    const float* bcol = B + (size_t)(ntile * 16 + ml);

    v8f c = {};
    for (int k = 0; k < K; k += 4) {
        v2f a = *reinterpret_cast<const v2f*>(arow + k);
        v2f b;
        b.x = bcol[(size_t)(k + hi * 2)     * N];
        b.y = bcol[(size_t)(k + hi * 2 + 1) * N];
        c = __builtin_amdgcn_wmma_f32_16x16x4_f32(
                /*neg_a=*/false, a, /*neg_b=*/false, b,
                /*c_mod=*/(short)0, c, /*reuse_a=*/false, /*reuse_b=*/false);
    }
    // C/D: vgpr i -> row (hi*8 + i), col = lane&15
    float* crow = C + (size_t)(mtile * 16 + hi * 8) * N + ntile * 16 + ml;
#pragma unroll
    for (int i = 0; i < 8; ++i) crow[(size_t)i * N] = c[i];
}

// ---------- attention coefficients: el/er[n,h] = <feat[n,h,:], a_l/r[h,:]> ----------
__global__ void attn_coef(const float* __restrict__ feat,
                          const float* __restrict__ al,
                          const float* __restrict__ ar,
                          float* __restrict__ el, float* __restrict__ er,
                          int H, int D) {
    int i = blockIdx.x * blockDim.x + threadIdx.x;   // i = n*H + h
    if (i >= N_NODES * H) return;
    int h = i % H;
    const float* f = feat + (size_t)i * D;
    const float* pl = al + h * D;
    const float* pr = ar + h * D;
    float sl = 0.f, sr = 0.f;
    for (int d = 0; d < D; ++d) { float v = f[d]; sl += v * pl[d]; sr += v * pr[d]; }
    el[i] = sl; er[i] = sr;
}

// ---------- fills ----------
__global__ void fill_u32(unsigned* p, unsigned v, int n) {
    int i = blockIdx.x * blockDim.x + threadIdx.x;
    if (i < n) p[i] = v;
}
__global__ void fill_f32(float* p, float v, int n) {
    int i = blockIdx.x * blockDim.x + threadIdx.x;
    if (i < n) p[i] = v;
}
__global__ void init_bias(float* __restrict__ out, const float* __restrict__ b,
                          int F, int total) {
    int i = blockIdx.x * blockDim.x + threadIdx.x;
    if (i < total) out[i] = b[i % F];
}

// ---------- edge softmax pass 1: segment max over dst ----------
__global__ void edge_max(const float* __restrict__ el, const float* __restrict__ er,
                         const int* __restrict__ src, const int* __restrict__ dst,
                         unsigned* __restrict__ menc, int H) {
    int e = blockIdx.x * blockDim.x + threadIdx.x;
    if (e >= N_EDGES) return;
    int s = src[e], d = dst[e];
    for (int h = 0; h < H; ++h) {
        float v = el[s * H + h] + er[d * H + h];
        v = v > 0.f ? v : NEG_SLOPE * v;
        atomicMax(&menc[d * H + h], fenc(v));
    }
}

// ---------- edge softmax pass 2: z = segment_sum(exp(e - m)) ----------
__global__ void edge_expsum(const float* __restrict__ el, const float* __restrict__ er,
                            const int* __restrict__ src, const int* __restrict__ dst,
                            const unsigned* __restrict__ menc, float* __restrict__ z,
                            int H) {
    int e = blockIdx.x * blockDim.x + threadIdx.x;
    if (e >= N_EDGES) return;
    int s = src[e], d = dst[e];
    for (int h = 0; h < H; ++h) {
        float v = el[s * H + h] + er[d * H + h];
        v = v > 0.f ? v : NEG_SLOPE * v;
        atomicAdd(&z[d * H + h], expf(v - fdec(menc[d * H + h])));
    }
}

// ---------- aggregate: out[dst,h,:] += feat[src,h,:] * alpha  (wave per edge) ----------
__global__ void edge_aggregate(const float* __restrict__ feat,
                               const float* __restrict__ el, const float* __restrict__ er,
                               const int* __restrict__ src, const int* __restrict__ dst,
                               const unsigned* __restrict__ menc, const float* __restrict__ z,
                               float* __restrict__ out, int H, int D) {
    int gwave = (int)((blockIdx.x * blockDim.x + threadIdx.x) >> 5);
    int lane  = threadIdx.x & 31;
    if (gwave >= N_EDGES) return;
    int s = src[gwave], d = dst[gwave];
    for (int h = 0; h < H; ++h) {
        float v = el[s * H + h] + er[d * H + h];
        v = v > 0.f ? v : NEG_SLOPE * v;
        float alpha = expf(v - fdec(menc[d * H + h])) / z[d * H + h];
        // D == 64: each lane owns 2 consecutive channels
        const float2 f = *reinterpret_cast<const float2*>(
            feat + ((size_t)s * H + h) * D + lane * 2);
        float* o = out + ((size_t)d * H + h) * D + lane * 2;
        atomicAdd(o,     f.x * alpha);
        atomicAdd(o + 1, f.y * alpha);
    }
}

// ---------- ELU in place ----------
__global__ void elu_inplace(float* __restrict__ x, int n) {
    int i = blockIdx.x * blockDim.x + threadIdx.x;
    if (i < n) {
        float v = x[i];
        x[i] = v > 0.f ? v : expm1f(v);
    }
}

extern "C" void kernel_launch(void* const* d_in, const int* in_sizes, int n_in,
                              void* d_out, int out_size, void* d_ws, size_t ws_size,
                              hipStream_t stream) {
    const float* x   = (const float*)d_in[0];
    const int*   src = (const int*)  d_in[1];
    const int*   dst = (const int*)  d_in[2];
    const float* W1  = (const float*)d_in[3];
    const float* al1 = (const float*)d_in[4];
    const float* ar1 = (const float*)d_in[5];
    const float* b1  = (const float*)d_in[6];
    const float* W2  = (const float*)d_in[7];
    const float* al2 = (const float*)d_in[8];
    const float* ar2 = (const float*)d_in[9];
    const float* b2  = (const float*)d_in[10];
    float* out = (float*)d_out;

    const int F1 = H1_HEADS * HID_D;           // 512
    // ---- workspace carve-up (floats) ----
    float*    ws    = (float*)d_ws;
    float*    feat1 = ws;                                   // [N,512]
    float*    hbuf  = feat1 + (size_t)N_NODES * F1;         // [N,512]
    float*    el1   = hbuf  + (size_t)N_NODES * F1;         // [N,8]
    float*    er1   = el1 + N_NODES * H1_HEADS;
    float*    z1    = er1 + N_NODES * H1_HEADS;
    unsigned* m1    = (unsigned*)(z1 + N_NODES * H1_HEADS);
    float*    el2   = (float*)(m1 + N_NODES * H1_HEADS);    // [N,1]
    float*    er2   = el2 + N_NODES;
    float*    z2    = er2 + N_NODES;
    unsigned* m2    = (unsigned*)(z2 + N_NODES);
    float*    feat2 = feat1;   // reuse: feat1 dead after layer-1 aggregation

    const int B = 256;
    auto blk = [](int n, int b) { return (n + b - 1) / b; };

    // ================= Layer 1 =================
    gemm_wmma_f32<<<dim3(N_NODES / 16, F1 / 64), 128, 0, stream>>>(x, W1, feat1, F_IN, F1);
    attn_coef<<<blk(N_NODES * H1_HEADS, B), B, 0, stream>>>(feat1, al1, ar1, el1, er1,
                                                            H1_HEADS, HID_D);
    fill_u32<<<blk(N_NODES * H1_HEADS, B), B, 0, stream>>>(m1, ENC_NEG_INF, N_NODES * H1_HEADS);
    fill_f32<<<blk(N_NODES * H1_HEADS, B), B, 0, stream>>>(z1, 0.f, N_NODES * H1_HEADS);
    edge_max<<<blk(N_EDGES, B), B, 0, stream>>>(el1, er1, src, dst, m1, H1_HEADS);
    edge_expsum<<<blk(N_EDGES, B), B, 0, stream>>>(el1, er1, src, dst, m1, z1, H1_HEADS);
    init_bias<<<blk(N_NODES * F1, B), B, 0, stream>>>(hbuf, b1, F1, N_NODES * F1);
    edge_aggregate<<<blk(N_EDGES * 32, B), B, 0, stream>>>(feat1, el1, er1, src, dst,
                                                           m1, z1, hbuf, H1_HEADS, HID_D);
    elu_inplace<<<blk(N_NODES * F1, B), B, 0, stream>>>(hbuf, N_NODES * F1);

    // ================= Layer 2 =================
    gemm_wmma_f32<<<dim3(N_NODES / 16, OUT_D / 64), 128, 0, stream>>>(hbuf, W2, feat2, F1, OUT_D);
    attn_coef<<<blk(N_NODES, B), B, 0, stream>>>(feat2, al2, ar2, el2, er2, 1, OUT_D);
    fill_u32<<<blk(N_NODES, B), B, 0, stream>>>(m2, ENC_NEG_INF, N_NODES);
    fill_f32<<<blk(N_NODES, B), B, 0, stream>>>(z2, 0.f, N_NODES);
    edge_max<<<blk(N_EDGES, B), B, 0, stream>>>(el2, er2, src, dst, m2, 1);
    edge_expsum<<<blk(N_EDGES, B), B, 0, stream>>>(el2, er2, src, dst, m2, z2, 1);
    init_bias<<<blk(N_NODES * OUT_D, B), B, 0, stream>>>(out, b2, OUT_D, N_NODES * OUT_D);
    edge_aggregate<<<blk(N_EDGES * 32, B), B, 0, stream>>>(feat2, el2, er2, src, dst,
                                                           m2, z2, out, 1, OUT_D);
    // H2 == 1: mean over heads is identity; `out` already holds the result.
}